// AlphaLayer_2448131358792
// MI455X (gfx1250) — compile-verified
//
#include <hip/hip_runtime.h>
#include <stdint.h>

// ---------------- problem/config constants ----------------
#define N_TOT 4194304            // 2^22 rows
#define CH 32                    // rows per thread-chunk
#define TPB 128                  // threads per block (kernels 1/3/5)
#define ROWS_PER_BLK (CH * TPB)  // 4096 rows per block
#define NBLK (N_TOT / ROWS_PER_BLK)  // 1024 blocks
#define NCHUNK (N_TOT / CH)          // 131072 chunks
#define SCAN_T 1024              // threads in scan kernels (2/4)
#define CPT (NCHUNK / SCAN_T)    // 128 chunks per scan thread

// workspace layout (float offsets)
#define WS_COMP 0                       // NCHUNK * 8 : forward composites
#define WS_STATE (NCHUNK * 8)           // NCHUNK * 2 : (cp,dp) entering each chunk
#define WS_BCOMP (WS_STATE + NCHUNK*2)  // NCHUNK * 2 : backward affine composites
#define WS_XIN (WS_BCOMP + NCHUNK*2)    // NCHUNK     : x entering each chunk from right

// LDS layouts: both arrays use the TDM-padded stride-33 layout (1 pad dword per 32),
// which is bank-conflict-free under wave32 (odd stride => distinct banks per lane).
#define ALDS 4226                // fidx(ROWS_PER_BLK+1) + 1  (alpha halo [s-1 .. s+R])
#define FLDS (TPB * (CH + 1))    // 4224 floats for f

__device__ __forceinline__ int fidx(int k) { return k + (k >> 5); } // pad 1 dword / 32

// ---------------- 7-parameter forward composite ----------------
// Row map on homogeneous (u,v,w): M = [[0,0,c],[0,-a,f],[-a,0,b]]
// Products keep shape [[p00,0,p02],[p10,pA,p12],[p20,0,p22]].
struct FComp { float p00, p02, p10, p12, p20, p22, pA; };

__device__ __forceinline__ FComp fIdentity() {
  FComp c; c.p00 = 1.f; c.p02 = 0.f; c.p10 = 0.f; c.p12 = 0.f;
  c.p20 = 0.f; c.p22 = 1.f; c.pA = 1.f; return c;
}
__device__ __forceinline__ FComp fCombine(const FComp& B, const FComp& A) { // B after A
  FComp r;
  r.p00 = B.p00 * A.p00 + B.p02 * A.p20;
  r.p02 = B.p00 * A.p02 + B.p02 * A.p22;
  r.p20 = B.p20 * A.p00 + B.p22 * A.p20;
  r.p22 = B.p20 * A.p02 + B.p22 * A.p22;
  r.p10 = B.p10 * A.p00 + B.pA  * A.p10 + B.p12 * A.p20;
  r.p12 = B.p10 * A.p02 + B.pA  * A.p12 + B.p12 * A.p22;
  r.pA  = B.pA * A.pA;
  return r;
}
__device__ __forceinline__ void fNorm(FComp& c) {
  float r = 1.0f / c.p22;  // diagonal dominance keeps p22 > 0
  c.p00 *= r; c.p02 *= r; c.p10 *= r; c.p12 *= r; c.p20 *= r; c.pA *= r;
  c.p22 = 1.0f;
}
// acc <- M(a,b,c,f) o acc  (specialized sparse product)
__device__ __forceinline__ void fStep(FComp& q, float a, float b, float c, float f) {
  FComp p;
  p.p00 = c * q.p20;
  p.p02 = c * q.p22;
  p.p10 = f * q.p20 - a * q.p10;
  p.p12 = f * q.p22 - a * q.p12;
  p.p20 = b * q.p20 - a * q.p00;
  p.p22 = b * q.p22 - a * q.p02;
  p.pA  = -a * q.pA;
  q = p;
}
__device__ __forceinline__ FComp load7(const float* p) {
  FComp c; c.p00 = p[0]; c.p02 = p[1]; c.p10 = p[2]; c.p12 = p[3];
  c.p20 = p[4]; c.p22 = p[5]; c.pA = p[6]; return c;
}
__device__ __forceinline__ void store7(float* p, const FComp& c) {
  p[0] = c.p00; p[1] = c.p02; p[2] = c.p10; p[3] = c.p12;
  p[4] = c.p20; p[5] = c.p22; p[6] = c.pA; p[7] = 0.f;
}

// ---------------- input staging: two TDM loads (f and alpha-with-halo) ----------------
#if defined(__gfx1250__) && __has_builtin(__builtin_amdgcn_tensor_load_to_lds) && \
    __has_builtin(__builtin_amdgcn_s_wait_tensorcnt)
#define USE_TDM 1
typedef unsigned int u32x4 __attribute__((ext_vector_type(4)));
typedef int i32x8 __attribute__((ext_vector_type(8)));
typedef int i32x4 __attribute__((ext_vector_type(4)));

// 1D f32 tile -> LDS, padded 1 dword per 32 dwords (stride-33 layout, matches fidx()).
// tensorLen < tileLen makes the TDM zero-fill the tail via its OOB rule.
__device__ __forceinline__ void tdm_load_1d(const float* gsrc, float* ldst,
                                            int tileLen, int tensorLen) {
  uint64_t ga = (uint64_t)(uintptr_t)(const void*)gsrc;
  uint32_t lds_off = (uint32_t)(uintptr_t)(void*)ldst;  // low 32 bits = LDS byte offset
  u32x4 g0;
  g0[0] = 1u;                                    // count=1 (valid descriptor)
  g0[1] = lds_off;                               // lds_addr
  g0[2] = (uint32_t)ga;                          // global_addr[31:0]
  g0[3] = (uint32_t)((ga >> 32) & 0x01FFFFFFu) | (2u << 30); // addr[56:32] | type=2
  i32x8 g1;
  g1[0] = (int)((2u << 16) |    // data_size = 4B
                (1u << 20) |    // pad_enable
                (4u << 22));    // pad_interval: 16x8B = 32 dwords; pad_amount=0 (1 dword)
  g1[1] = (int)(((unsigned)tensorLen & 0xFFFFu) << 16);               // tensor_dim0 lo
  g1[2] = (int)((((unsigned)tensorLen >> 16) & 0xFFFFu) | (1u << 16)); // dim0 hi | dim1=1
  g1[3] = (int)(((unsigned)tileLen & 0xFFFFu) << 16);                 // tile_dim0
  g1[4] = 1;                                                          // tile_dim1 = 1
  g1[5] = tensorLen;                                                  // dim0_stride lo
  g1[6] = 0; g1[7] = 0;
  i32x4 gz4; gz4[0] = 0; gz4[1] = 0; gz4[2] = 0; gz4[3] = 0;  // groups 2/3 unused (2D)
  i32x8 gz8; gz8[0] = 0; gz8[1] = 0; gz8[2] = 0; gz8[3] = 0;
  gz8[4] = 0; gz8[5] = 0; gz8[6] = 0; gz8[7] = 0;
  __builtin_amdgcn_tensor_load_to_lds(g0, g1, gz4, gz4, gz8, 0);
}
#endif

__device__ __forceinline__ void stage_inputs(const float* __restrict__ alpha,
                                             const float* __restrict__ fg,
                                             float* la, float* lf,
                                             int s, int tid) {
  const int gbase = (s == 0) ? 0 : s - 1;                 // alpha halo base
  const int atensor = (N_TOT - gbase < ROWS_PER_BLK + 2)  // zero-fill past end of alpha
                          ? (N_TOT - gbase) : (ROWS_PER_BLK + 2);
#ifdef USE_TDM
  const bool wave0 = (__builtin_amdgcn_readfirstlane((int)tid) < 32);
  if (wave0) {
    tdm_load_1d(fg + s, lf, ROWS_PER_BLK, ROWS_PER_BLK);
    tdm_load_1d(alpha + gbase, la, ROWS_PER_BLK + 2, atensor);
    __builtin_amdgcn_s_wait_tensorcnt(0);  // both DMAs retired
  }
#else
  for (int k = tid; k < ROWS_PER_BLK; k += TPB) lf[fidx(k)] = fg[s + k];
  for (int k = tid; k < ROWS_PER_BLK + 2; k += TPB)
    la[fidx(k)] = (k < atensor) ? alpha[gbase + k] : 0.0f;
#endif
}

// coefficients for global row i; kb = LDS index of alpha[i-1] in the halo layout
__device__ __forceinline__ void coefs(const float* la, int kb, int i,
                                      float& a, float& b, float& c) {
  float al = la[fidx(kb < 0 ? 0 : kb)];  // alpha[i-1] (kb<0 only at i==0, value unused)
  float am = la[fidx(kb + 1)];           // alpha[i]
  float ar = la[fidx(kb + 2)];           // alpha[i+1]
  a = al * al;
  b = 1.0f + am * am * am;
  c = (i + 1 < N_TOT) ? (ar * ar + 2.0f * ar) : 0.0f;
}

// ---------------- Kernel 1: per-chunk forward composites ----------------
__global__ __launch_bounds__(TPB) void k_fwd_compose(const float* __restrict__ alpha,
                                                     const float* __restrict__ fg,
                                                     float* __restrict__ ws) {
  __shared__ float la[ALDS];
  __shared__ float lf[FLDS];
  const int tid = threadIdx.x;
  const int s = blockIdx.x * ROWS_PER_BLK;
  stage_inputs(alpha, fg, la, lf, s, tid);
  __syncthreads();

  const int cbase = tid * CH;
  const int kadj = (s == 0) ? -1 : 0;  // halo base shift for block 0
  FComp acc = fIdentity();
  for (int j = 0; j < CH; ++j) {
    int i = s + cbase + j;
    float a, b, c;
    coefs(la, cbase + j + kadj, i, a, b, c);
    float fv = lf[fidx(cbase + j)];
    if (i == 0) {  // init: t0 = (c0, f0, b0), independent of input state
      acc.p00 = 0.f; acc.p02 = c; acc.p10 = 0.f; acc.p12 = fv;
      acc.p20 = 0.f; acc.p22 = b; acc.pA = 0.f;
    } else {
      fStep(acc, a, b, c, fv);
    }
  }
  fNorm(acc);
  int k = blockIdx.x * TPB + tid;
  store7(ws + WS_COMP + (size_t)k * 8, acc);
}

// ---------------- Kernel 2: single-WG scan of all forward composites ----------------
__global__ __launch_bounds__(SCAN_T) void k_fwd_scan(float* __restrict__ ws) {
  __shared__ FComp sc[SCAN_T];
  const int t = threadIdx.x;
  FComp acc = fIdentity();
  for (int j = 0; j < CPT; ++j) {
    const float* p = ws + WS_COMP + (size_t)(t * CPT + j) * 8;
    __builtin_prefetch((const void*)(p + 64), 0, 1);
    acc = fCombine(load7(p), acc);
    fNorm(acc);
  }
  sc[t] = acc;
  __syncthreads();
  for (int off = 1; off < SCAN_T; off <<= 1) {  // inclusive Hillis-Steele
    FComp v = sc[t];
    FComp u = (t >= off) ? sc[t - off] : fIdentity();
    __syncthreads();
    FComp w = fCombine(v, u);
    fNorm(w);
    sc[t] = w;
    __syncthreads();
  }
  FComp pre = (t > 0) ? sc[t - 1] : fIdentity();  // exclusive prefix
  for (int j = 0; j < CPT; ++j) {
    int k = t * CPT + j;
    float inv = 1.0f / pre.p22;
    ws[WS_STATE + 2 * k]     = pre.p02 * inv;  // cp entering chunk k
    ws[WS_STATE + 2 * k + 1] = pre.p12 * inv;  // dp entering chunk k
    pre = fCombine(load7(ws + WS_COMP + (size_t)k * 8), pre);
    fNorm(pre);
  }
}

// ---------------- Kernel 3: per-chunk backward affine composites ----------------
__global__ __launch_bounds__(TPB) void k_bwd_compose(const float* __restrict__ alpha,
                                                     const float* __restrict__ fg,
                                                     float* __restrict__ ws) {
  __shared__ float la[ALDS];
  __shared__ float lf[FLDS];
  const int tid = threadIdx.x;
  const int s = blockIdx.x * ROWS_PER_BLK;
  stage_inputs(alpha, fg, la, lf, s, tid);
  __syncthreads();

  const int cbase = tid * CH;
  const int kadj = (s == 0) ? -1 : 0;
  const int k = blockIdx.x * TPB + tid;
  float cp = ws[WS_STATE + 2 * k];
  float dp = ws[WS_STATE + 2 * k + 1];
  float bs = 1.0f, bo = 0.0f;  // affine composite x -> bs*x + bo
  for (int j = 0; j < CH; ++j) {
    int i = s + cbase + j;
    float a, b, c;
    coefs(la, cbase + j + kadj, i, a, b, c);
    float fv = lf[fidx(cbase + j)];
    if (i == 0) {
      float inv = 1.0f / b; cp = c * inv; dp = fv * inv;
    } else {
      float m = b - a * cp;
      float inv = 1.0f / m;
      cp = c * inv;
      dp = (fv - a * dp) * inv;
    }
    bo = bs * dp + bo;   // acc = acc o (x -> dp - cp*x)
    bs = bs * (-cp);
  }
  ws[WS_BCOMP + 2 * k]     = bs;
  ws[WS_BCOMP + 2 * k + 1] = bo;
}

// ---------------- Kernel 4: single-WG suffix scan of backward composites ----------------
__global__ __launch_bounds__(SCAN_T) void k_bwd_scan(float* __restrict__ ws) {
  __shared__ float ss[SCAN_T];
  __shared__ float so[SCAN_T];
  const int t = threadIdx.x;
  float as = 1.0f, ao = 0.0f;
  for (int j = 0; j < CPT; ++j) {
    int k = t * CPT + j;
    __builtin_prefetch((const void*)(ws + WS_BCOMP + 2 * k + 64), 0, 1);
    float hs = ws[WS_BCOMP + 2 * k], ho = ws[WS_BCOMP + 2 * k + 1];
    ao = as * ho + ao;  // acc = acc o H_k
    as = as * hs;
  }
  ss[t] = as; so[t] = ao;
  __syncthreads();
  for (int off = 1; off < SCAN_T; off <<= 1) {  // inclusive suffix scan
    float vs = ss[t], vo = so[t];
    float us = 1.0f, uo = 0.0f;
    if (t + off < SCAN_T) { us = ss[t + off]; uo = so[t + off]; }
    __syncthreads();
    so[t] = vs * uo + vo;  // v o u
    ss[t] = vs * us;
    __syncthreads();
  }
  float So = (t < SCAN_T - 1) ? so[t + 1] : 0.0f;  // exclusive suffix applied to 0
  float v = So;
  for (int j = CPT - 1; j >= 0; --j) {
    int k = t * CPT + j;
    ws[WS_XIN + k] = v;
    float hs = ws[WS_BCOMP + 2 * k], ho = ws[WS_BCOMP + 2 * k + 1];
    v = hs * v + ho;
  }
}

// ---------------- Kernel 5: apply — recompute cp/dp, back-substitute, store x ----------------
__global__ __launch_bounds__(TPB) void k_apply(const float* __restrict__ alpha,
                                               const float* __restrict__ fg,
                                               const float* __restrict__ ws,
                                               float* __restrict__ out) {
  __shared__ float la[ALDS];
  __shared__ float lf[FLDS];
  const int tid = threadIdx.x;
  const int s = blockIdx.x * ROWS_PER_BLK;
  stage_inputs(alpha, fg, la, lf, s, tid);
  __syncthreads();

  const int cbase = tid * CH;
  const int kadj = (s == 0) ? -1 : 0;
  const int k = blockIdx.x * TPB + tid;
  float cp = ws[WS_STATE + 2 * k];
  float dp = ws[WS_STATE + 2 * k + 1];
  // forward pass: cp_j kept in a fully-unrolled register array (alpha LDS is shared
  // between neighboring chunks now, so it must not be overwritten); dp_j overwrites
  // the thread-private f slot.
  float cpv[CH];
#pragma unroll
  for (int j = 0; j < CH; ++j) {
    int i = s + cbase + j;
    float a, b, c;
    coefs(la, cbase + j + kadj, i, a, b, c);
    float fv = lf[fidx(cbase + j)];
    if (i == 0) {
      float inv = 1.0f / b; cp = c * inv; dp = fv * inv;
    } else {
      float m = b - a * cp;
      float inv = 1.0f / m;
      cp = c * inv;
      dp = (fv - a * dp) * inv;
    }
    cpv[j] = cp;
    lf[fidx(cbase + j)] = dp;
  }
  // backward within chunk
  float v = ws[WS_XIN + k];
#pragma unroll
  for (int j = CH - 1; j >= 0; --j) {
    float dj = lf[fidx(cbase + j)];
    float x = dj - cpv[j] * v;
    lf[fidx(cbase + j)] = x;     // reuse f slots to hold x
    v = x;
  }
  __syncthreads();
  // coalesced store of the block's 4096 results
  for (int kk = tid; kk < ROWS_PER_BLK; kk += TPB) out[s + kk] = lf[fidx(kk)];
}

// ---------------- launch ----------------
extern "C" void kernel_launch(void* const* d_in, const int* in_sizes, int n_in,
                              void* d_out, int out_size, void* d_ws, size_t ws_size,
                              hipStream_t stream) {
  (void)in_sizes; (void)n_in; (void)out_size; (void)ws_size;
  const float* alpha = (const float*)d_in[0];
  const float* f     = (const float*)d_in[1];
  float* out = (float*)d_out;
  float* ws  = (float*)d_ws;   // needs ~6.6 MB

  k_fwd_compose<<<NBLK, TPB, 0, stream>>>(alpha, f, ws);
  k_fwd_scan<<<1, SCAN_T, 0, stream>>>(ws);
  k_bwd_compose<<<NBLK, TPB, 0, stream>>>(alpha, f, ws);
  k_bwd_scan<<<1, SCAN_T, 0, stream>>>(ws);
  k_apply<<<NBLK, TPB, 0, stream>>>(alpha, f, ws, out);
}